// GINEEncoder_20169166422332
// MI455X (gfx1250) — compile-verified
//
#include <hip/hip_runtime.h>

// GINE encoder for MI455X (gfx1250, wave32).
// Memory-bound workload (≈3 GB HBM traffic vs 38 GFLOP of GEMM) -> keep fp32
// fidelity and use V_WMMA_F32_16X16X4_F32 for all dense GEMMs.
// GEMM kernel: 32x128 tile per block, W staged in LDS (K-paired layout -> B
// fragment = 1 conflict-free ds_load_b64), 2 accumulator chains per wave.

#define CH      128   // node channel count (IN_CH == OUT_CH == 128)
#define EDIM    16    // edge attr dim
#define MTILE   32    // rows per block
#define ASTRIDE 132   // LDS dword stride for A rows (conflict-free b64 reads)
#define WPSTRIDE 288  // LDS dword stride per K-pair of W (skews lane halves to
                      // disjoint 32-bank groups; 64 pairs * 288 * 4B = 72 KB)
#define SMEM_DWORDS (MTILE * ASTRIDE + 64 * WPSTRIDE)

typedef __attribute__((ext_vector_type(2))) float v2f;
typedef __attribute__((ext_vector_type(8))) float v8f;

// Out[r, c] = relu( In[r, :] @ W[:, c] + bias[c] )
// where In[r, k] = Agg ? (1+eps)*A[r,k] + Agg[r,k] : A[r,k]   (GINE combine fused)
// Block: 256 threads = 8 waves. Wave w owns cols [16w,16w+16) for both 16-row
// subtiles. K loop: 32 steps x 2 x v_wmma_f32_16x16x4_f32 (B reused across subtiles).
// In-place safe (A==Out): block stages its 32 global rows into LDS and syncs
// before any global store; no other block touches those rows.
__global__ __launch_bounds__(256) void node_gemm_relu(
    const float* __restrict__ A, const float* __restrict__ Agg,
    const float* __restrict__ epsPtr,
    const float* __restrict__ W, const float* __restrict__ bias,
    float* __restrict__ Out)
{
    extern __shared__ float smem[];
    float* sA = smem;                    // MTILE x ASTRIDE
    float* sW = smem + MTILE * ASTRIDE;  // 64 pairs x WPSTRIDE: {W[2p][c], W[2p+1][c]}

    const int  tid = threadIdx.x;
    const long r0  = (long)blockIdx.x * MTILE;

    float scale = 1.0f;
    if (epsPtr) scale = 1.0f + epsPtr[0];          // wave-uniform scalar load

    // Stage A tile (coalesced b32), GINE combine fused.
    #pragma unroll
    for (int j = 0; j < (MTILE * CH) / 256; ++j) {     // 16 iters
        const int  lin = tid + j * 256;
        const int  row = lin >> 7;
        const int  col = lin & 127;
        const long g   = (r0 + row) * CH + col;
        float v = A[g];
        if (Agg) v = scale * v + Agg[g];
        sA[row * ASTRIDE + col] = v;
    }
    // Stage W (coalesced b128 reads) into K-paired layout.
    #pragma unroll
    for (int j = 0; j < (CH * CH) / (256 * 4); ++j) {  // 16 iters of float4
        const int    i4 = tid + j * 256;               // 0..4095
        const float4 wv = ((const float4*)W)[i4];
        const int    d0 = i4 << 2;                     // flat dword index
        const int    k  = d0 >> 7;                     // source row of W
        const int    c0 = d0 & 127;                    // first of 4 columns
        float* dst = sW + (k >> 1) * WPSTRIDE + (k & 1);
        dst[2 * (c0 + 0)] = wv.x;
        dst[2 * (c0 + 1)] = wv.y;
        dst[2 * (c0 + 2)] = wv.z;
        dst[2 * (c0 + 3)] = wv.w;
    }
    __syncthreads();

    const int wave = tid >> 5;
    const int lane = tid & 31;
    const int half = lane >> 4;    // lane group: selects K pair {4ks,4ks+1} vs {4ks+2,4ks+3}
    const int lm   = lane & 15;    // M (for A) / N (for B,C,D)
    const int cb   = wave * 16;    // this wave's output column base

    const float* aRow0 = sA + lm * ASTRIDE;
    const float* aRow1 = sA + (16 + lm) * ASTRIDE;
    const float* bCol  = sW + 2 * (cb + lm) + half * WPSTRIDE;

    v8f c0 = {};
    v8f c1 = {};
    #pragma unroll
    for (int ks = 0; ks < 32; ++ks) {
        const int k0 = ks * 4 + 2 * half;
        // B fragment (4x16 f32): one conflict-free ds_load_b64, reused twice.
        v2f b  = *(const v2f*)(bCol + 2 * ks * WPSTRIDE);
        // A fragments for the two 16-row subtiles.
        v2f a0 = *(const v2f*)(aRow0 + k0);
        v2f a1 = *(const v2f*)(aRow1 + k0);
        c0 = __builtin_amdgcn_wmma_f32_16x16x4_f32(false, a0, false, b,
                                                   (short)0, c0, false, false);
        c1 = __builtin_amdgcn_wmma_f32_16x16x4_f32(false, a1, false, b,
                                                   (short)0, c1, false, false);
    }

    // Epilogue: bias + ReLU. C/D layout: VGPR r -> row r (lanes 0-15) / r+8 (16-31).
    const float bb = bias[cb + lm];
    #pragma unroll
    for (int r = 0; r < 8; ++r) {
        const long m = r0 + r + 8 * half;
        float v0 = c0[r] + bb;
        float v1 = c1[r] + bb;
        v0 = v0 > 0.0f ? v0 : 0.0f;
        v1 = v1 > 0.0f ? v1 : 0.0f;
        Out[m * CH + cb + lm]        = v0;
        Out[(m + 16) * CH + cb + lm] = v1;
    }
}

// Per edge e: msg = relu(h[src] + (edge_attr[e] @ We + be)); agg[dst] += msg.
// One wave per edge (wave32): lane handles 4 channels (float4 => 512 B/row coalesced).
__global__ __launch_bounds__(256) void edge_msg_scatter(
    const float* __restrict__ h, const int* __restrict__ ei,
    const float* __restrict__ ea, const float* __restrict__ We,
    const float* __restrict__ be, float* __restrict__ agg, int nEdges)
{
    __shared__ float sWe[EDIM * CH];   // 8 KB
    __shared__ float sbe[CH];

    const int tid = threadIdx.x;
    for (int j = tid; j < EDIM * CH; j += 256) sWe[j] = We[j];
    if (tid < CH) sbe[tid] = be[tid];
    __syncthreads();

    const int e = blockIdx.x * 8 + (tid >> 5);
    if (e >= nEdges) return;
    const int lane = tid & 31;

    const long src = ei[e];
    const long dst = ei[(long)nEdges + e];

    // Edge linear: 4 output channels per lane, K=16 in registers.
    float a0 = sbe[lane * 4 + 0];
    float a1 = sbe[lane * 4 + 1];
    float a2 = sbe[lane * 4 + 2];
    float a3 = sbe[lane * 4 + 3];
    const float* eav = ea + (long)e * EDIM;        // wave-uniform 64 B
    #pragma unroll
    for (int k = 0; k < EDIM; ++k) {
        const float  ek = eav[k];
        const float4 w  = ((const float4*)(sWe + k * CH))[lane];
        a0 = fmaf(ek, w.x, a0);
        a1 = fmaf(ek, w.y, a1);
        a2 = fmaf(ek, w.z, a2);
        a3 = fmaf(ek, w.w, a3);
    }

    // Gather h[src] (coalesced 512 B row), message ReLU, scatter-add.
    const float4 hv = *(const float4*)(h + src * CH + lane * 4);
    float m0 = hv.x + a0; m0 = m0 > 0.0f ? m0 : 0.0f;
    float m1 = hv.y + a1; m1 = m1 > 0.0f ? m1 : 0.0f;
    float m2 = hv.z + a2; m2 = m2 > 0.0f ? m2 : 0.0f;
    float m3 = hv.w + a3; m3 = m3 > 0.0f ? m3 : 0.0f;

    float* ag = agg + dst * CH + lane * 4;
    atomicAdd(ag + 0, m0);
    atomicAdd(ag + 1, m1);
    atomicAdd(ag + 2, m2);
    atomicAdd(ag + 3, m3);
}

extern "C" void kernel_launch(void* const* d_in, const int* in_sizes, int n_in,
                              void* d_out, int out_size, void* d_ws, size_t ws_size,
                              hipStream_t stream)
{
    const float* x    = (const float*)d_in[0];
    const int*   ei   = (const int*)  d_in[1];   // [2, E] int32
    const float* ea   = (const float*)d_in[2];
    const float* Wenc = (const float*)d_in[3];
    const float* benc = (const float*)d_in[4];
    const float* We1  = (const float*)d_in[5];
    const float* be1  = (const float*)d_in[6];
    const float* W11  = (const float*)d_in[7];
    const float* b11  = (const float*)d_in[8];
    const float* W12  = (const float*)d_in[9];
    const float* b12  = (const float*)d_in[10];
    const float* eps1 = (const float*)d_in[11];
    const float* We2  = (const float*)d_in[12];
    const float* be2  = (const float*)d_in[13];
    const float* W21  = (const float*)d_in[14];
    const float* b21  = (const float*)d_in[15];
    const float* W22  = (const float*)d_in[16];
    const float* b22  = (const float*)d_in[17];
    const float* eps2 = (const float*)d_in[18];

    const int N = in_sizes[0] / CH;     // 200000 (divisible by 32)
    const int E = in_sizes[2] / EDIM;   // 600000

    float* h   = (float*)d_ws;                    // N*CH floats
    float* agg = h + (size_t)N * CH;              // N*CH floats
    float* t   = (float*)d_out;                   // MLP intermediate lives in d_out

    const int    gemmBlocks = N / MTILE;          // 6250
    const int    edgeBlocks = (E + 7) / 8;        // 75000
    const size_t smemBytes  = SMEM_DWORDS * sizeof(float);   // ~90.6 KB dynamic LDS

    // h = relu(x @ W_enc + b_enc)
    node_gemm_relu<<<gemmBlocks, 256, smemBytes, stream>>>(x, nullptr, nullptr, Wenc, benc, h);

    // ---- conv1 ----
    hipMemsetAsync(agg, 0, (size_t)N * CH * sizeof(float), stream);
    edge_msg_scatter<<<edgeBlocks, 256, 0, stream>>>(h, ei, ea, We1, be1, agg, E);
    node_gemm_relu<<<gemmBlocks, 256, smemBytes, stream>>>(h, agg, eps1, W11, b11, t);
    node_gemm_relu<<<gemmBlocks, 256, smemBytes, stream>>>(t, nullptr, nullptr, W12, b12, h);

    // ---- conv2 ----
    hipMemsetAsync(agg, 0, (size_t)N * CH * sizeof(float), stream);
    edge_msg_scatter<<<edgeBlocks, 256, 0, stream>>>(h, ei, ea, We2, be2, agg, E);
    node_gemm_relu<<<gemmBlocks, 256, smemBytes, stream>>>(h, agg, eps2, W21, b21, t);
    // Final layer in-place on d_out (block stages A-tile in LDS before storing).
    node_gemm_relu<<<gemmBlocks, 256, smemBytes, stream>>>(t, nullptr, nullptr, W22, b22, (float*)d_out);
}